// RangeBN_41437844471847
// MI455X (gfx1250) — compile-verified
//
#include <hip/hip_runtime.h>
#include <math.h>

// Clang ext-vectors (needed for WMMA operands and nontemporal b128 stores;
// HIP's float4 is a struct, not a vector type).
typedef float f2 __attribute__((ext_vector_type(2)));
typedef float f4 __attribute__((ext_vector_type(4)));
typedef float f8 __attribute__((ext_vector_type(8)));

#define B_N   20
#define C_N   256
#define HW_N  4096
#define CHW_N (C_N * HW_N)        // 1,048,576 per sample
#define NROWS (B_N * HW_N)        // 81,920
#define NCHUNK 10
#define KPAD   12                 // chunks padded to multiple of 4 for WMMA K
#define QMAXF 255.0f

// ---- workspace layout (32-bit word offsets into d_ws) ----
#define OFF_SMIN 0      // 20 x u32 (ordered-encoded per-sample min)
#define OFF_SMAX 32     // 20 x u32
#define OFF_SCAL 64     // [0]=scale_x [1]=inv_scale_x [2]=zp_x
#define OFF_QW   128    // 256 f32 quantized weight
#define OFF_QB   384    // 256 f32 quantized bias
#define OFF_CMIN 640    // 12*256 f32 raw min  per (chunk, channel), k=10,11 zeroed
#define OFF_CMAX 3712   // 12*256 f32 raw max
#define OFF_CSUM 6784   // 12*256 f32 sum(fq(x))
#define OFF_COEF 9856   // 256 f32
#define OFF_OFFS 10112  // 256 f32   (total: 10368 words = 41,472 bytes)

// ---- ordered-float <-> uint encoding for integer min/max atomics ----
__device__ __forceinline__ unsigned enc_f(float f) {
  unsigned u = __float_as_uint(f);
  return (u & 0x80000000u) ? ~u : (u | 0x80000000u);
}
__device__ __forceinline__ float dec_f(unsigned e) {
  return __uint_as_float((e & 0x80000000u) ? (e ^ 0x80000000u) : ~e);
}

// ---- wave32 reductions ----
__device__ __forceinline__ float wredMin(float v) {
  for (int o = 16; o; o >>= 1) v = fminf(v, __shfl_xor(v, o, 32));
  return v;
}
__device__ __forceinline__ float wredMax(float v) {
  for (int o = 16; o; o >>= 1) v = fmaxf(v, __shfl_xor(v, o, 32));
  return v;
}
__device__ __forceinline__ float wredSum(float v) {
  for (int o = 16; o; o >>= 1) v += __shfl_xor(v, o, 32);
  return v;
}

// fake-quant forward value: (rint(clip(x*inv_s + zp, 0, 255)) - zp) * s
__device__ __forceinline__ float fqx(float v, float s, float is, float zp) {
  float t = fminf(fmaxf(__builtin_fmaf(v, is, zp), 0.0f), QMAXF);
  return (rintf(t) - zp) * s;
}

// ---------------- K0: init atomic targets + zero K-padding slots ----------------
__global__ __launch_bounds__(512) void rbn_k0_init(unsigned* __restrict__ wsU) {
  int t = threadIdx.x;  // 512 threads
  if (t < B_N) {
    wsU[OFF_SMIN + t] = 0xFFFFFFFFu;  // encoded +inf (min identity)
    wsU[OFF_SMAX + t] = 0u;           // encoded -inf (max identity)
  }
  // zero pad chunks k = 10, 11 (512 entries per array):
  // cmin = cmax = 0 -> fq(0)-fq(0) == 0 range contribution; csum pad = 0.
  float* wsF = (float*)wsU;
  wsF[OFF_CMIN + NCHUNK * C_N + t] = 0.0f;
  wsF[OFF_CMAX + NCHUNK * C_N + t] = 0.0f;
  wsF[OFF_CSUM + NCHUNK * C_N + t] = 0.0f;
}

// ---------------- K1: per-sample min/max (streaming, prefetch) ----------------
__global__ __launch_bounds__(256) void rbn_k1_sample_minmax(
    const float* __restrict__ x, unsigned* __restrict__ wsU) {
  const int BLKS = 64;                     // blocks per sample
  int b   = blockIdx.x / BLKS;
  int blk = blockIdx.x % BLKS;
  const int PER = CHW_N / 4 / BLKS;        // 4096 f4 per block
  const f4* xp = (const f4*)(x + (size_t)b * CHW_N) + (size_t)blk * PER;

  float vmin = __builtin_inff(), vmax = -__builtin_inff();
  for (int j = threadIdx.x; j < PER; j += 256) {
    __builtin_prefetch((const void*)(xp + j + 256), 0, 0);  // global_prefetch_b8
    f4 v = xp[j];
    vmin = fminf(vmin, fminf(fminf(v.x, v.y), fminf(v.z, v.w)));
    vmax = fmaxf(vmax, fmaxf(fmaxf(v.x, v.y), fmaxf(v.z, v.w)));
  }
  vmin = wredMin(vmin);
  vmax = wredMax(vmax);

  __shared__ float sMin[8], sMax[8];
  int lane = threadIdx.x & 31, wid = threadIdx.x >> 5;
  if (lane == 0) { sMin[wid] = vmin; sMax[wid] = vmax; }
  __syncthreads();
  if (threadIdx.x == 0) {
    float m = sMin[0], M = sMax[0];
    for (int i = 1; i < 8; ++i) { m = fminf(m, sMin[i]); M = fmaxf(M, sMax[i]); }
    atomicMin(&wsU[OFF_SMIN + b], enc_f(m));
    atomicMax(&wsU[OFF_SMAX + b], enc_f(M));
  }
}

// ---------------- K1b: scalar quant params + qweight + qbias ----------------
__global__ __launch_bounds__(256) void rbn_k1b_scalars(
    const float* __restrict__ weight, const float* __restrict__ bias,
    float* __restrict__ wsF) {
  const unsigned* wsU = (const unsigned*)wsF;
  __shared__ float sh[24];
  __shared__ float scal[3];
  int t = threadIdx.x, lane = t & 31, wid = t >> 5;

  float wv = weight[t];
  float bv = bias[t];
  float wmn = wredMin(wv), wmx = wredMax(wv), bs = wredSum(bv);
  if (lane == 0) { sh[wid] = wmn; sh[8 + wid] = wmx; sh[16 + wid] = bs; }
  __syncthreads();

  if (t == 0) {
    float wmin = sh[0], wmax = sh[8], bsum = sh[16];
    for (int i = 1; i < 8; ++i) {
      wmin = fminf(wmin, sh[i]); wmax = fmaxf(wmax, sh[8 + i]); bsum += sh[16 + i];
    }
    // batch-mean of per-sample min / max
    float mnsum = 0.f, mxsum = 0.f;
    for (int b = 0; b < B_N; ++b) {
      mnsum += dec_f(wsU[OFF_SMIN + b]);
      mxsum += dec_f(wsU[OFF_SMAX + b]);
    }
    float min_v = mnsum * (1.0f / B_N), max_v = mxsum * (1.0f / B_N);
    float sx = (max_v - min_v) / QMAXF;
    if (sx == 0.0f) sx = 1.0f;
    float zx = truncf(fminf(fmaxf(-min_v / sx, 0.0f), QMAXF));
    wsF[OFF_SCAL + 0] = sx;
    wsF[OFF_SCAL + 1] = 1.0f / sx;
    wsF[OFF_SCAL + 2] = zx;

    float sw = (wmax - wmin) / QMAXF;
    if (sw == 0.0f) sw = 1.0f;
    float zw = truncf(fminf(fmaxf(-wmin / sw, 0.0f), QMAXF));
    float bmean = bsum * (1.0f / C_N);
    float zb = truncf(fminf(fmaxf(-bmean, 0.0f), QMAXF));  // qbias path: scale==0 -> 1
    scal[0] = sw; scal[1] = zw; scal[2] = zb;
  }
  __syncthreads();

  float sw = scal[0], zw = scal[1], zb = scal[2];
  wsF[OFF_QW + t] = (rintf(fminf(fmaxf(wv / sw + zw, 0.0f), QMAXF)) - zw) * sw;
  wsF[OFF_QB + t] = rintf(fminf(fmaxf(bv + zb, 0.0f), QMAXF)) - zb;
}

// ---------------- K2: per-(chunk, channel) raw min/max + sum(fq(x)) ----------------
// One block per (k, c): two contiguous 16 KB runs (batches 2k and 2k+1). x is
// L2-resident after K1 (84 MB < 192 MB L2), so this pass reads from L2.
__global__ __launch_bounds__(256) void rbn_k2_chunk_stats(
    const float* __restrict__ x, float* __restrict__ wsF) {
  int k = blockIdx.x >> 8;
  int c = blockIdx.x & 255;
  float sx = wsF[OFF_SCAL + 0], is = wsF[OFF_SCAL + 1], zx = wsF[OFF_SCAL + 2];

  float vmin = __builtin_inff(), vmax = -__builtin_inff(), vsum = 0.f;
#pragma unroll
  for (int r = 0; r < 2; ++r) {
    const f4* p = (const f4*)(x + ((size_t)(2 * k + r) * C_N + c) * HW_N);
#pragma unroll
    for (int j = 0; j < 4; ++j) {
      f4 v = p[threadIdx.x + j * 256];
      vmin = fminf(vmin, fminf(fminf(v.x, v.y), fminf(v.z, v.w)));
      vmax = fmaxf(vmax, fmaxf(fmaxf(v.x, v.y), fmaxf(v.z, v.w)));
      vsum += fqx(v.x, sx, is, zx) + fqx(v.y, sx, is, zx) +
              fqx(v.z, sx, is, zx) + fqx(v.w, sx, is, zx);
    }
  }
  vmin = wredMin(vmin); vmax = wredMax(vmax); vsum = wredSum(vsum);

  __shared__ float sMin[8], sMax[8], sSum[8];
  int lane = threadIdx.x & 31, wid = threadIdx.x >> 5;
  if (lane == 0) { sMin[wid] = vmin; sMax[wid] = vmax; sSum[wid] = vsum; }
  __syncthreads();
  if (threadIdx.x == 0) {
    float m = sMin[0], M = sMax[0], S = sSum[0];
    for (int i = 1; i < 8; ++i) {
      m = fminf(m, sMin[i]); M = fmaxf(M, sMax[i]); S += sSum[i];
    }
    wsF[OFF_CMIN + blockIdx.x] = m;
    wsF[OFF_CMAX + blockIdx.x] = M;
    wsF[OFF_CSUM + blockIdx.x] = S;
  }
}

// ---------------- K3: per-channel coef/offset via WMMA row-sums ----------------
// Csum[c] = sum_k range(k,c), chanSum[c] = sum_k csum2(k,c): both are
// A[16ch x 12] * ones row-sums -> 3x accumulating V_WMMA_F32_16X16X4_F32.
// Chunk arrays are padded to K=12 with zeros so every load is unconditional:
// no EXEC manipulation anywhere around the WMMAs (ISA: EXEC must be all 1s).
__device__ __forceinline__ float rangeAt(const float* __restrict__ wsF, int k,
                                         int c, float sx, float is, float zx) {
  float mn = wsF[OFF_CMIN + k * C_N + c];
  float mx = wsF[OFF_CMAX + k * C_N + c];
  return fqx(mx, sx, is, zx) - fqx(mn, sx, is, zx);  // fq is monotone
}

__global__ __launch_bounds__(512) void rbn_k3_channel_params(float* __restrict__ wsF) {
  int t = threadIdx.x;
  int lane = t & 31, w = t >> 5;       // 16 waves, 16 channels per wave
  int m = lane & 15, kh = lane >> 4;   // A 16x4 f32 layout: lane<16 -> K0..1, lane>=16 -> K2..3
  int chbase = w * 16;
  int c = chbase + m;

  float sx = wsF[OFF_SCAL + 0], is = wsF[OFF_SCAL + 1], zx = wsF[OFF_SCAL + 2];

  f8 accR = {};  // sum over chunks of range(k, c)
  f8 accS = {};  // sum over chunks of partial channel sums
  f2 bOnes = {1.0f, 1.0f};

#pragma unroll
  for (int kk = 0; kk < 3; ++kk) {
    int k0 = kk * 4 + 2 * kh;          // lanes 0-15: K0..1, lanes 16-31: K2..3
    f2 aR, aS;
    aR.x = rangeAt(wsF, k0, c, sx, is, zx);
    aR.y = rangeAt(wsF, k0 + 1, c, sx, is, zx);
    aS.x = wsF[OFF_CSUM + k0 * C_N + c];
    aS.y = wsF[OFF_CSUM + (k0 + 1) * C_N + c];
    accR = __builtin_amdgcn_wmma_f32_16x16x4_f32(false, aR, false, bOnes,
                                                 (short)0, accR, false, false);
    accS = __builtin_amdgcn_wmma_f32_16x16x4_f32(false, aS, false, bOnes,
                                                 (short)0, accS, false, false);
  }

  // D layout: VGPR j of lanes 0-15 holds row j; lanes 16-31 hold row j+8.
  if (m == 0) {
    float cst = 0.5f * 0.35f *
                (1.0f + sqrtf(3.14159265358979f * logf(4.0f))) /
                sqrtf(2.0f * logf((float)(NROWS / NCHUNK)));
#pragma unroll
    for (int j = 0; j < 8; ++j) {
      int ch = chbase + kh * 8 + j;
      float Csum = accR[j] * cst;
      float mean = accS[j] * (1.0f / (float)NROWS);
      float scl  = 1.0f / (Csum * (1.0f / NCHUNK) + 1e-7f);
      float coef = scl * wsF[OFF_QW + ch];
      wsF[OFF_COEF + ch] = coef;
      wsF[OFF_OFFS + ch] = wsF[OFF_QB + ch] - mean * coef;
    }
  }
}

// ---------------- K4: out = fq(x)*coef[c] + offset[c] (NT stores) ----------------
// One block per (b, c) slab (4096 contiguous floats) -> coef/offset are
// wave-uniform scalar loads. Non-temporal stores keep x resident in L2.
__global__ __launch_bounds__(256) void rbn_k4_output(
    const float* __restrict__ x, float* __restrict__ out,
    const float* __restrict__ wsF) {
  int slab = blockIdx.x;          // b*C + c
  int c = slab & 255;
  float sx = wsF[OFF_SCAL + 0], is = wsF[OFF_SCAL + 1], zx = wsF[OFF_SCAL + 2];
  float coef = wsF[OFF_COEF + c];
  float offs = wsF[OFF_OFFS + c];

  const f4* xp = (const f4*)(x + (size_t)slab * HW_N);
  f4* op = (f4*)(out + (size_t)slab * HW_N);
#pragma unroll
  for (int j = 0; j < 4; ++j) {
    int i = threadIdx.x + j * 256;
    f4 v = xp[i];
    f4 o;
    o.x = __builtin_fmaf(fqx(v.x, sx, is, zx), coef, offs);
    o.y = __builtin_fmaf(fqx(v.y, sx, is, zx), coef, offs);
    o.z = __builtin_fmaf(fqx(v.z, sx, is, zx), coef, offs);
    o.w = __builtin_fmaf(fqx(v.w, sx, is, zx), coef, offs);
    __builtin_nontemporal_store(o, &op[i]);
  }
}

extern "C" void kernel_launch(void* const* d_in, const int* in_sizes, int n_in,
                              void* d_out, int out_size, void* d_ws, size_t ws_size,
                              hipStream_t stream) {
  (void)in_sizes; (void)n_in; (void)out_size; (void)ws_size;
  const float* x      = (const float*)d_in[0];
  const float* weight = (const float*)d_in[1];
  const float* bias   = (const float*)d_in[2];
  float* out = (float*)d_out;
  float* wsF = (float*)d_ws;
  unsigned* wsU = (unsigned*)d_ws;

  rbn_k0_init<<<1, 512, 0, stream>>>(wsU);
  rbn_k1_sample_minmax<<<B_N * 64, 256, 0, stream>>>(x, wsU);
  rbn_k1b_scalars<<<1, 256, 0, stream>>>(weight, bias, wsF);
  rbn_k2_chunk_stats<<<NCHUNK * C_N, 256, 0, stream>>>(x, wsF);
  rbn_k3_channel_params<<<1, 512, 0, stream>>>(wsF);
  rbn_k4_output<<<B_N * C_N, 256, 0, stream>>>(x, out, wsF);
}